// CrossAttention_27255862460837
// MI455X (gfx1250) — compile-verified
//
#include <hip/hip_runtime.h>

typedef __attribute__((ext_vector_type(2))) float v2f;
typedef __attribute__((ext_vector_type(8))) float v8f;

#define BATCH 6
#define NTOK  4096
#define QDIM  320
#define CTXN  77
#define CTXD  768
#define NH    8
#define DH    40
#define BH    (BATCH * NH)   // 48
#define DHP   48             // DH padded to 3x16
#define CNP   80             // CTXN padded to 5x16
#define SCALE_F 0.15811388300841897f  // 40^-0.5

__device__ __forceinline__ v8f wmma_f32(v2f a, v2f b, v8f c) {
  // D = A(16x4) * B(4x16) + C(16x16), full fp32
  return __builtin_amdgcn_wmma_f32_16x16x4_f32(false, a, false, b, (short)0, c,
                                               false, false);
}

// ---------------------------------------------------------------------------
// Kernel 1: K/V projections with zero padding.
//   KpT[bh][d(48)][j(80)] = (context @ Wk) transposed per head (zero padded)
//   Vp [bh][j(80)][d(48)] = (context @ Wv) per head            (zero padded)
// ---------------------------------------------------------------------------
__global__ void kv_proj_kernel(const float* __restrict__ ctx,
                               const float* __restrict__ Wk,
                               const float* __restrict__ Wv,
                               float* __restrict__ KpT,
                               float* __restrict__ Vp) {
  int idx = blockIdx.x * blockDim.x + threadIdx.x;
  const int per = DHP * CNP;          // 3840 per bh
  const int totK = BH * per;
  if (idx < totK) {
    int bh = idx / per, r = idx % per;
    int d = r / CNP, j = r % CNP;
    float acc = 0.f;
    if (d < DH && j < CTXN) {
      int b = bh >> 3, h = bh & 7;
      const float* c = ctx + (size_t)(b * CTXN + j) * CTXD;
      const float* w = Wk + h * DH + d;           // Wk[c][n], row stride QDIM
      for (int cc = 0; cc < CTXD; ++cc) acc += c[cc] * w[(size_t)cc * QDIM];
    }
    KpT[idx] = acc;
  } else if (idx < 2 * totK) {
    idx -= totK;
    int bh = idx / per, r = idx % per;
    int j = r / DHP, d = r % DHP;
    float acc = 0.f;
    if (d < DH && j < CTXN) {
      int b = bh >> 3, h = bh & 7;
      const float* c = ctx + (size_t)(b * CTXN + j) * CTXD;
      const float* w = Wv + h * DH + d;
      for (int cc = 0; cc < CTXD; ++cc) acc += c[cc] * w[(size_t)cc * QDIM];
    }
    Vp[idx] = acc;
  }
}

// ---------------------------------------------------------------------------
// Kernel 2: Q projection (WMMA), 64x32 register tile per wave
// (4 M-subtiles x 2 N-subtiles, 8 accumulators). K = 320.
// Qp[bh][i][d(48 pad)] = SCALE * (x @ Wq)
// ---------------------------------------------------------------------------
__global__ void q_proj_kernel(const float* __restrict__ x,
                              const float* __restrict__ Wq,
                              float* __restrict__ Qp) {
  int wv   = blockIdx.x * (blockDim.x >> 5) + (threadIdx.x >> 5);
  int lane = threadIdx.x & 31;
  int l16  = lane & 15;
  int hi   = lane >> 4;                 // K half select (+2)
  const int NT = QDIM / 32;             // 10 column tiles of 32
  int mt = wv / NT, nt = wv % NT;
  int m0 = mt * 64, n0 = nt * 32;

  const float* xrow[4];
#pragma unroll
  for (int s = 0; s < 4; ++s)
    xrow[s] = x + (size_t)(m0 + s * 16 + l16) * QDIM;
  int col0 = n0 + l16, col1 = n0 + 16 + l16;

  v8f zero = {0.f, 0.f, 0.f, 0.f, 0.f, 0.f, 0.f, 0.f};
  v8f c[4][2];
#pragma unroll
  for (int s = 0; s < 4; ++s) { c[s][0] = zero; c[s][1] = zero; }

  for (int kk = 0; kk < QDIM; kk += 4) {
    int ka = kk + hi * 2;
    const float* w0 = Wq + (size_t)ka * QDIM;
    const float* w1 = w0 + QDIM;
    v2f b0; b0.x = w0[col0]; b0.y = w1[col0];
    v2f b1; b1.x = w0[col1]; b1.y = w1[col1];
    v2f a[4];
#pragma unroll
    for (int s = 0; s < 4; ++s) { a[s].x = xrow[s][ka]; a[s].y = xrow[s][ka + 1]; }
#pragma unroll
    for (int s = 0; s < 4; ++s) {
      c[s][0] = wmma_f32(a[s], b0, c[s][0]);
      c[s][1] = wmma_f32(a[s], b1, c[s][1]);
    }
  }

#pragma unroll
  for (int t = 0; t < 2; ++t) {
    int col = n0 + t * 16 + l16;
    int h = col / DH, d = col % DH;
#pragma unroll
    for (int s = 0; s < 4; ++s) {
#pragma unroll
      for (int r = 0; r < 8; ++r) {
        int g = m0 + s * 16 + r + hi * 8;        // global token row
        int b = g >> 12, i = g & (NTOK - 1);     // NTOK = 4096
        Qp[((size_t)(b * NH + h) * NTOK + i) * DHP + d] = c[s][t][r] * SCALE_F;
      }
    }
  }
}

// ---------------------------------------------------------------------------
// Kernel 3: fused attention per (bh, 16-row tile):
//   sim = Qp @ KpT (WMMA, 5 col tiles) -> LDS
//   rear_sim store + softmax + reweight (in LDS)
//   out_mid = attn @ Vp (WMMA, 3 col tiles)
// ---------------------------------------------------------------------------
__global__ void attn_kernel(const float* __restrict__ Qp,
                            const float* __restrict__ KpT,
                            const float* __restrict__ Vp,
                            float* __restrict__ rear,
                            float* __restrict__ omid) {
  __shared__ float smem[4][16][CNP];         // 20 KB / block
  int wslot = threadIdx.x >> 5;
  int wv    = blockIdx.x * 4 + wslot;
  int lane  = threadIdx.x & 31;
  int l16   = lane & 15;
  int hi    = lane >> 4;
  int bh = wv >> 8;                          // / 256 row-tiles
  int it = wv & 255;
  int i0 = it * 16;
  float (*sm)[CNP] = smem[wslot];

  v8f zero = {0.f, 0.f, 0.f, 0.f, 0.f, 0.f, 0.f, 0.f};
  v8f acc[5];
#pragma unroll
  for (int t = 0; t < 5; ++t) acc[t] = zero;

  // Phase A: sim tile (16 x 80) via WMMA, K = 48 (padded head dim)
  const float* qrow = Qp + ((size_t)bh * NTOK + i0 + l16) * DHP;
  for (int kk = 0; kk < DHP; kk += 4) {
    int ka = kk + hi * 2;
    v2f a; a.x = qrow[ka]; a.y = qrow[ka + 1];
    const float* kb0 = KpT + ((size_t)bh * DHP + ka) * CNP;
    const float* kb1 = kb0 + CNP;
#pragma unroll
    for (int t = 0; t < 5; ++t) {
      int j = t * 16 + l16;
      v2f bb; bb.x = kb0[j]; bb.y = kb1[j];
      acc[t] = wmma_f32(a, bb, acc[t]);
    }
  }
#pragma unroll
  for (int t = 0; t < 5; ++t)
#pragma unroll
    for (int r = 0; r < 8; ++r)
      sm[r + hi * 8][t * 16 + l16] = acc[t][r];
  __syncthreads();

  // Phase B: rear_sim store, softmax, cross-attention-control reweight
  if (lane < 16) {
    int i = i0 + lane;
    float* row = sm[lane];
    float mx = -3.0e38f;
    for (int j = 0; j < CTXN; ++j) mx = fmaxf(mx, row[j]);
    float* rs = rear + ((size_t)bh * NTOK + i) * CTXN;
    float sum = 0.f;
    for (int j = 0; j < CTXN; ++j) {
      float s = row[j];
      rs[j] = s;                              // pre-softmax (scaled) sim
      float e = expf(s - mx);
      row[j] = e;
      sum += e;
    }
    float inv = 1.f / sum;
    for (int j = 0; j < CTXN; ++j) row[j] *= inv;
    int b = bh >> 3;
    if (b >= 3) {                             // reweight token 2, batches 3..5
      const float RW[3] = {-2.f, 1.f, 5.f};
      row[2] *= RW[b - 3];
    }
    row[77] = 0.f; row[78] = 0.f; row[79] = 0.f;   // pad cols
  }
  __syncthreads();

  // Phase C: out_mid tile (16 x 48) = attn(16x80, LDS) @ Vp(80x48)
  v8f o[3];
#pragma unroll
  for (int t = 0; t < 3; ++t) o[t] = zero;
  const float* arow = &sm[l16][0];
  for (int kk = 0; kk < CNP; kk += 4) {
    int ka = kk + hi * 2;
    v2f a; a.x = arow[ka]; a.y = arow[ka + 1];
    const float* vb0 = Vp + ((size_t)bh * CNP + ka) * DHP;
    const float* vb1 = vb0 + DHP;
#pragma unroll
    for (int t = 0; t < 3; ++t) {
      int d = t * 16 + l16;
      v2f bb; bb.x = vb0[d]; bb.y = vb1[d];
      o[t] = wmma_f32(a, bb, o[t]);
    }
  }
  int b = bh >> 3, h = bh & 7;
#pragma unroll
  for (int t = 0; t < 3; ++t) {
    int d = t * 16 + l16;
    if (d < DH) {
#pragma unroll
      for (int r = 0; r < 8; ++r) {
        int i = i0 + r + hi * 8;
        omid[((size_t)b * NTOK + i) * QDIM + h * DH + d] = o[t][r];
      }
    }
  }
}

// ---------------------------------------------------------------------------
// Kernel 4: final projection out = out_mid @ Wo + bo (WMMA),
// 64x32 register tile per wave (8 accumulators).
// ---------------------------------------------------------------------------
__global__ void out_proj_kernel(const float* __restrict__ omid,
                                const float* __restrict__ Wo,
                                const float* __restrict__ bo,
                                float* __restrict__ out) {
  int wv   = blockIdx.x * (blockDim.x >> 5) + (threadIdx.x >> 5);
  int lane = threadIdx.x & 31;
  int l16  = lane & 15;
  int hi   = lane >> 4;
  const int NT = QDIM / 32;             // 10 column tiles of 32
  int mt = wv / NT, nt = wv % NT;
  int m0 = mt * 64, n0 = nt * 32;

  const float* arow[4];
#pragma unroll
  for (int s = 0; s < 4; ++s)
    arow[s] = omid + (size_t)(m0 + s * 16 + l16) * QDIM;
  int col0 = n0 + l16, col1 = n0 + 16 + l16;

  v8f zero = {0.f, 0.f, 0.f, 0.f, 0.f, 0.f, 0.f, 0.f};
  v8f c[4][2];
#pragma unroll
  for (int s = 0; s < 4; ++s) { c[s][0] = zero; c[s][1] = zero; }

  for (int kk = 0; kk < QDIM; kk += 4) {
    int ka = kk + hi * 2;
    const float* w0 = Wo + (size_t)ka * QDIM;
    const float* w1 = w0 + QDIM;
    v2f b0; b0.x = w0[col0]; b0.y = w1[col0];
    v2f b1; b1.x = w0[col1]; b1.y = w1[col1];
    v2f a[4];
#pragma unroll
    for (int s = 0; s < 4; ++s) { a[s].x = arow[s][ka]; a[s].y = arow[s][ka + 1]; }
#pragma unroll
    for (int s = 0; s < 4; ++s) {
      c[s][0] = wmma_f32(a[s], b0, c[s][0]);
      c[s][1] = wmma_f32(a[s], b1, c[s][1]);
    }
  }

  float bias0 = bo[col0];
  float bias1 = bo[col1];
#pragma unroll
  for (int s = 0; s < 4; ++s) {
#pragma unroll
    for (int r = 0; r < 8; ++r) {
      int g = m0 + s * 16 + r + hi * 8;
      out[(size_t)g * QDIM + col0] = c[s][0][r] + bias0;
      out[(size_t)g * QDIM + col1] = c[s][1][r] + bias1;
    }
  }
}

// ---------------------------------------------------------------------------
extern "C" void kernel_launch(void* const* d_in, const int* in_sizes, int n_in,
                              void* d_out, int out_size, void* d_ws,
                              size_t ws_size, hipStream_t stream) {
  const float* x   = (const float*)d_in[0];
  const float* ctx = (const float*)d_in[1];
  const float* Wq  = (const float*)d_in[2];
  const float* Wk  = (const float*)d_in[3];
  const float* Wv  = (const float*)d_in[4];
  const float* Wo  = (const float*)d_in[5];
  const float* bo  = (const float*)d_in[6];
  // d_in[7] = time, unused (reweight always applies for time >= 0)

  float* out  = (float*)d_out;                         // [B][N][320]
  float* rear = out + (size_t)BATCH * NTOK * QDIM;     // [B][H][N][77]

  float* ws   = (float*)d_ws;
  float* Qp   = ws;                                    // 48*4096*48
  float* KpT  = Qp  + (size_t)BH * NTOK * DHP;         // 48*48*80
  float* Vp   = KpT + (size_t)BH * DHP * CNP;          // 48*80*48
  float* omid = Vp  + (size_t)BH * CNP * DHP;          // 24576*320

  // 1) K/V projections (padded, zeros in pad region)
  int tot1 = 2 * BH * DHP * CNP;                       // 368640
  kv_proj_kernel<<<(tot1 + 255) / 256, 256, 0, stream>>>(ctx, Wk, Wv, KpT, Vp);

  // 2) Q projection: (24576/64)*(320/32) = 3840 waves, 4 waves/block
  q_proj_kernel<<<3840 / 4, 128, 0, stream>>>(x, Wq, Qp);

  // 3) Fused attention: 48*256 = 12288 wave-tiles, 4 waves/block
  attn_kernel<<<(BH * 256) / 4, 128, 0, stream>>>(Qp, KpT, Vp, rear, omid);

  // 4) Output projection: 3840 waves, 4 waves/block
  out_proj_kernel<<<3840 / 4, 128, 0, stream>>>(omid, Wo, bo, out);
}